// VocabEmbeddingWithLoRA_6588479832425
// MI455X (gfx1250) — compile-verified
//
#include <hip/hip_runtime.h>

// ---------------------------------------------------------------------------
// VocabEmbeddingWithLoRA for MI455X (gfx1250, wave32)
//
//   out[t, d] = base_weight[x[t], d] + sum_r lora_A[r, x[t]] * lora_B[d, r]
//
// Memory-bound: ~280 MB total traffic -> ~12us floor @ 23.3 TB/s.
// LoRA GEMM (M=16 tokens, K=16 rank, N=2048) done on the WMMA pipe with
// V_WMMA_F32_16X16X4_F32 (4 chained K-steps per 16x16 tile, A reused).
// lora_A (8 MB) and lora_B (128 KB) are L2-resident on a 192 MB L2.
// ---------------------------------------------------------------------------

typedef __attribute__((ext_vector_type(2))) float v2f;
typedef __attribute__((ext_vector_type(8))) float v8f;

#define VOCAB   128000
#define D_MODEL 2048
#define RANK    16
#define TILE_M  16          // tokens per block (one WMMA M-tile)
#define WAVES   8           // 256 threads / wave32
#define NT_PER_WAVE (D_MODEL / 16 / WAVES)   // 16 N-tiles of 16 cols per wave

__global__ __launch_bounds__(256)
void embed_lora_wmma_kernel(const int*   __restrict__ x,     // [NTOK]
                            const float* __restrict__ W,     // [VOCAB, D_MODEL]
                            const float* __restrict__ A,     // [RANK, VOCAB]
                            const float* __restrict__ Bm,    // [D_MODEL, RANK]
                            float*       __restrict__ out)   // [NTOK, D_MODEL]
{
    __shared__ int toks[TILE_M];

    const int tile = blockIdx.x;            // 16-token tile index
    const int lane = threadIdx.x & 31;
    const int wave = threadIdx.x >> 5;

    if (threadIdx.x < TILE_M)
        toks[threadIdx.x] = x[tile * TILE_M + threadIdx.x];
    __syncthreads();

    // ---- A fragment (16x4 F32 per WMMA, ISA layout):
    //   lanes 0-15 : M=lane,    VGPR0=K0, VGPR1=K1
    //   lanes 16-31: M=lane-16, VGPR0=K2, VGPR1=K3
    // ar[m][k] = lora_A[k, tok[m]]  (column gather; lora_A stays in L2)
    const int m_a   = lane & 15;
    const int khalf = (lane >> 4) * 2;       // 0 or 2
    const int tok_a = toks[m_a];
    v2f afrag[4];
#pragma unroll
    for (int kk = 0; kk < 4; ++kk) {
        const int k0 = kk * 4 + khalf;
        afrag[kk].x = A[(size_t)(k0    ) * VOCAB + tok_a];
        afrag[kk].y = A[(size_t)(k0 + 1) * VOCAB + tok_a];
    }

    // ---- C/D layout: lane L holds column N = L&15; VGPR v holds row
    //   M = v + (L>=16 ? 8 : 0). Precompute the 8 row bases this lane touches.
    const int mrow_base = (lane >> 4) * 8;
    size_t wrow[8], orow[8];
#pragma unroll
    for (int v = 0; v < 8; ++v) {
        const int mr = mrow_base + v;
        wrow[v] = (size_t)toks[mr] * D_MODEL;                       // gather row
        orow[v] = ((size_t)tile * TILE_M + mr) * (size_t)D_MODEL;   // output row
    }

    const int n = lane & 15;

    for (int nt = wave * NT_PER_WAVE; nt < (wave + 1) * NT_PER_WAVE; ++nt) {
        const int d = nt * 16 + n;           // this lane's output column

        // Issue the base-embedding gather early so it overlaps the WMMA chain.
        float basev[8];
#pragma unroll
        for (int v = 0; v < 8; ++v)
            basev[v] = W[wrow[v] + d];

        // ---- B fragments (4x16 F32 per WMMA): B[k][n] = lora_B[d, k].
        //   lanes 0-15 : VGPR0=K0 row, VGPR1=K1 row
        //   lanes 16-31: VGPR0=K2 row, VGPR1=K3 row
        // Consecutive k -> one 8-byte (global_load_b64) load per K-step.
        v8f c = {};
#pragma unroll
        for (int kk = 0; kk < 4; ++kk) {
            const int kb = kk * 4 + khalf;
            const v2f b = *reinterpret_cast<const v2f*>(
                Bm + (size_t)d * RANK + kb);
            // D = A(16x4) * B(4x16) + C ; accumulate across the 4 K-steps.
            c = __builtin_amdgcn_wmma_f32_16x16x4_f32(
                    /*neg_a=*/false, afrag[kk],
                    /*neg_b=*/false, b,
                    /*c_mod=*/(short)0, c,
                    /*reuse_a=*/false, /*reuse_b=*/false);
        }

        // ---- Fused epilogue: base + lora, stores hit two contiguous 64B
        // segments per instruction (lanes 0-15 row M=v, lanes 16-31 row M=v+8).
#pragma unroll
        for (int v = 0; v < 8; ++v)
            out[orow[v] + d] = basev[v] + c[v];
    }
}

extern "C" void kernel_launch(void* const* d_in, const int* in_sizes, int n_in,
                              void* d_out, int out_size, void* d_ws, size_t ws_size,
                              hipStream_t stream) {
    const int*   x  = (const int*)  d_in[0];   // token ids [B*S]
    const float* W  = (const float*)d_in[1];   // base_weight [VOCAB, D_MODEL]
    const float* A  = (const float*)d_in[2];   // lora_A [RANK, VOCAB]
    const float* Bm = (const float*)d_in[3];   // lora_B [D_MODEL, RANK]
    float*       o  = (float*)      d_out;     // [B*S, D_MODEL]

    const int ntok   = in_sizes[0];            // 16384
    const int blocks = ntok / TILE_M;          // 1024

    embed_lora_wmma_kernel<<<blocks, 256, 0, stream>>>(x, W, A, Bm, o);
}